// Att_60421599920797
// MI455X (gfx1250) — compile-verified
//
#include <hip/hip_runtime.h>
#include <hip/hip_bf16.h>
#include <math.h>

#define B_  2
#define T_  512
#define C_  2048
#define H_  32
#define HS_ 64
#define EPS_ 1e-5f

typedef __attribute__((ext_vector_type(2))) float v2f;
typedef __attribute__((ext_vector_type(8))) float v8f;
typedef __attribute__((ext_vector_type(4))) int   v4i;

#if __has_builtin(__builtin_amdgcn_global_load_async_to_lds_b128)
#define USE_ASYNC_LDS 1
typedef __attribute__((address_space(1))) v4i* gasp;   // global (AS1) int4*
typedef __attribute__((address_space(3))) v4i* lasp;   // LDS (AS3) int4*
// integer-cast route: always legal, strips const, applies target AS.
#define GCAST(p) ((gasp)(unsigned long long)(p))
#define LCAST(p) ((lasp)(unsigned)(unsigned long long)(p))
#else
#define USE_ASYNC_LDS 0
#endif

__device__ __forceinline__ void wait_async0() {
#if __has_builtin(__builtin_amdgcn_s_wait_asynccnt)
  __builtin_amdgcn_s_wait_asynccnt(0);
#else
  asm volatile("s_wait_asynccnt 0x0" ::: "memory");
#endif
}

// ---------------- kernel 1: token shift + 4-way time-mix ----------------
__global__ __launch_bounds__(256)
void mix_kernel(const float* __restrict__ x, const float* __restrict__ attstate,
                const float* __restrict__ mk, const float* __restrict__ mv,
                const float* __restrict__ mr, const float* __restrict__ mg,
                float* __restrict__ xk, float* __restrict__ xv,
                float* __restrict__ xr, float* __restrict__ xg,
                float* __restrict__ att_out) {
  int idx = blockIdx.x * blockDim.x + threadIdx.x;
  if (idx >= B_ * T_ * C_) return;
  int c = idx % C_;
  int t = (idx / C_) % T_;
  int b = idx / (T_ * C_);
  float xi = x[idx];
  float xp = (t == 0) ? attstate[b * C_ + c] : x[idx - C_];
  float k_ = mk[c], v_ = mv[c], r_ = mr[c], g_ = mg[c];
  xk[idx] = xi * k_ + xp * (1.f - k_);
  xv[idx] = xi * v_ + xp * (1.f - v_);
  xr[idx] = xi * r_ + xp * (1.f - r_);
  xg[idx] = xi * g_ + xp * (1.f - g_);
  if (t == T_ - 1) att_out[b * C_ + c] = xi;
}

__global__ void wdecay_kernel(const float* __restrict__ td, float* __restrict__ w) {
  int i = blockIdx.x * blockDim.x + threadIdx.x;
  if (i < H_ * HS_) w[i] = expf(-expf(td[i]));
}

// ---------------- WMMA fp32 GEMM: C[m][n] = sum_k A[m][k] * W[n][k] ----------------
// M = B_*T_ = 1024, N = C_ = 2048, K = C_ = 2048 (all multiples of tile sizes)
#define BM 64
#define BN 128
#define BK 32
#define LDSS 36          // 32 + 4 pad, keeps float4 (16B) row alignment
#define NCHUNK (C_ / BK)

struct GemmBatch {
  const float* A[4];
  const float* W[4];
  float*       O[4];
  int          mode[4];   // 0 plain, 1 silu
};

__global__ __launch_bounds__(256)
void gemm_wmma(GemmBatch gb) {
  const float* __restrict__ A    = gb.A[blockIdx.z];
  const float* __restrict__ W    = gb.W[blockIdx.z];
  float*       __restrict__ Cmat = gb.O[blockIdx.z];
  const int mode = gb.mode[blockIdx.z];

  __shared__ float As[2][BM * LDSS];
  __shared__ float Ws[2][BN * LDSS];

  const int K    = C_;
  const int tid  = threadIdx.x;
  const int m0   = blockIdx.y * BM;
  const int n0   = blockIdx.x * BN;
  const int wid  = tid >> 5;
  const int lane = tid & 31;
  const int wm   = (wid & 3) * 16;       // wave row offset within block tile
  const int wn   = (wid >> 2) * 64;      // wave col offset within block tile
  const int ml   = lane & 15;            // row/col within 16
  const int kh   = (lane >> 4) * 2;      // lanes 16-31 handle k+2,k+3

  // per-thread staging coordinates (float4 granularity)
  int arow[2], acol[2], wrow[4], wcol[4];
#pragma unroll
  for (int p = 0; p < 2; p++) {
    int i4 = tid + p * 256;
    arow[p] = i4 >> 3;
    acol[p] = (i4 & 7) << 2;
  }
#pragma unroll
  for (int p = 0; p < 4; p++) {
    int i4 = tid + p * 256;
    wrow[p] = i4 >> 3;
    wcol[p] = (i4 & 7) << 2;
  }

  v8f acc[4];
  const v8f vzero = {0.f, 0.f, 0.f, 0.f, 0.f, 0.f, 0.f, 0.f};
#pragma unroll
  for (int i = 0; i < 4; i++) acc[i] = vzero;

#if USE_ASYNC_LDS
  // --- async (ASYNCcnt-tracked) double-buffered pipeline ---
  {
    const int k0 = 0;
#pragma unroll
    for (int p = 0; p < 2; p++)
      __builtin_amdgcn_global_load_async_to_lds_b128(
          GCAST(&A[(size_t)(m0 + arow[p]) * K + k0 + acol[p]]),
          LCAST(&As[0][arow[p] * LDSS + acol[p]]), 0, 0);
#pragma unroll
    for (int p = 0; p < 4; p++)
      __builtin_amdgcn_global_load_async_to_lds_b128(
          GCAST(&W[(size_t)(n0 + wrow[p]) * K + k0 + wcol[p]]),
          LCAST(&Ws[0][wrow[p] * LDSS + wcol[p]]), 0, 0);
  }
  for (int ch = 0; ch < NCHUNK; ch++) {
    const int buf = ch & 1;
    wait_async0();          // our buf's transfers have landed
    __syncthreads();        // everyone's transfers landed; prev buf reads done
    if (ch + 1 < NCHUNK) {
      const int k0 = (ch + 1) * BK;
      const int nb = buf ^ 1;
#pragma unroll
      for (int p = 0; p < 2; p++)
        __builtin_amdgcn_global_load_async_to_lds_b128(
            GCAST(&A[(size_t)(m0 + arow[p]) * K + k0 + acol[p]]),
            LCAST(&As[nb][arow[p] * LDSS + acol[p]]), 0, 0);
#pragma unroll
      for (int p = 0; p < 4; p++)
        __builtin_amdgcn_global_load_async_to_lds_b128(
            GCAST(&W[(size_t)(n0 + wrow[p]) * K + k0 + wcol[p]]),
            LCAST(&Ws[nb][wrow[p] * LDSS + wcol[p]]), 0, 0);
    }
#pragma unroll
    for (int kk = 0; kk < BK; kk += 4) {
      v2f a = *(const v2f*)&As[buf][(wm + ml) * LDSS + kk + kh];
#pragma unroll
      for (int nt = 0; nt < 4; nt++) {
        v2f b = *(const v2f*)&Ws[buf][(wn + nt * 16 + ml) * LDSS + kk + kh];
        acc[nt] = __builtin_amdgcn_wmma_f32_16x16x4_f32(
            false, a, false, b, (short)0, acc[nt], false, false);
      }
    }
  }
#else
  // --- fallback: synchronous staging ---
  for (int ch = 0; ch < NCHUNK; ch++) {
    const int k0  = ch * BK;
    const int buf = ch & 1;
#pragma unroll
    for (int p = 0; p < 2; p++)
      *(float4*)&As[buf][arow[p] * LDSS + acol[p]] =
          *(const float4*)&A[(size_t)(m0 + arow[p]) * K + k0 + acol[p]];
#pragma unroll
    for (int p = 0; p < 4; p++)
      *(float4*)&Ws[buf][wrow[p] * LDSS + wcol[p]] =
          *(const float4*)&W[(size_t)(n0 + wrow[p]) * K + k0 + wcol[p]];
    __syncthreads();
#pragma unroll
    for (int kk = 0; kk < BK; kk += 4) {
      v2f a = *(const v2f*)&As[buf][(wm + ml) * LDSS + kk + kh];
#pragma unroll
      for (int nt = 0; nt < 4; nt++) {
        v2f b = *(const v2f*)&Ws[buf][(wn + nt * 16 + ml) * LDSS + kk + kh];
        acc[nt] = __builtin_amdgcn_wmma_f32_16x16x4_f32(
            false, a, false, b, (short)0, acc[nt], false, false);
      }
    }
    __syncthreads();
  }
#endif

  // D layout: VGPR v -> rows (v, v+8); lanes 0-15 cols ml, lanes 16-31 rows +8
  const int rbase = m0 + wm + ((lane >> 4) << 3);
  const int cbase = n0 + wn + ml;
#pragma unroll
  for (int nt = 0; nt < 4; nt++) {
#pragma unroll
    for (int v = 0; v < 8; v++) {
      float val = acc[nt][v];
      if (mode == 1) val = val / (1.f + expf(-val));  // silu
      Cmat[(size_t)(rbase + v) * C_ + cbase + nt * 16] = val;
    }
  }
}

// ---------------- WKV recurrent scan: one block per (b,h), one lane per column ----------------
__global__ __launch_bounds__(64)
void wkv_scan(const float* __restrict__ r, const float* __restrict__ k,
              const float* __restrict__ v, const float* __restrict__ wdec,
              const float* __restrict__ faaaa, const float* __restrict__ S0,
              float* __restrict__ out, float* __restrict__ Sout) {
  const int b = blockIdx.x / H_;
  const int h = blockIdx.x % H_;
  const int j = threadIdx.x;

  __shared__ float  f_lds[HS_];
  __shared__ float2 rk2[2][HS_];
  __shared__ float  rkdot[T_];

  f_lds[j] = faaaa[h * HS_ + j];

  float wreg[HS_];
#pragma unroll
  for (int i = 0; i < HS_; i++) wreg[i] = wdec[h * HS_ + i];

  float S[HS_];
#pragma unroll
  for (int i = 0; i < HS_; i++)
    S[i] = S0[((size_t)(b * H_ + h) * HS_ + i) * HS_ + j];
  __syncthreads();

  // precompute rkdot[t] = sum_i r[t,i]*faaaa[i]*k[t,i]  (8 timesteps per thread)
  for (int rep = 0; rep < T_ / HS_; rep++) {
    int t = j + rep * HS_;
    const float* rp = &r[(size_t)(b * T_ + t) * C_ + h * HS_];
    const float* kp = &k[(size_t)(b * T_ + t) * C_ + h * HS_];
    float acc = 0.f;
#pragma unroll
    for (int i = 0; i < HS_; i++) acc += rp[i] * f_lds[i] * kp[i];
    rkdot[t] = acc;
  }
  __syncthreads();

  for (int t = 0; t < T_; t++) {
    const size_t idx = (size_t)(b * T_ + t) * C_ + h * HS_ + j;
    float rj = r[idx], kj = k[idx], vj = v[idx];
    int buf = t & 1;
    rk2[buf][j] = make_float2(rj, kj);
    __syncthreads();
    float acc = rkdot[t] * vj;      // (r*faaaa) . k  contribution times v_j
#pragma unroll
    for (int i = 0; i < HS_; i++) {
      float2 rk = rk2[buf][i];      // LDS broadcast (same address across wave)
      acc += rk.x * S[i];           // r . S_old column j
      S[i] = S[i] * wreg[i] + rk.y * vj;  // S = S*w + outer(k,v)
    }
    out[idx] = acc;
  }

#pragma unroll
  for (int i = 0; i < HS_; i++)
    Sout[((size_t)(b * H_ + h) * HS_ + i) * HS_ + j] = S[i];
}

// ---------------- GroupNorm(/8) over HS per head + gate multiply ----------------
__global__ __launch_bounds__(256)
void gnorm_gate(const float* __restrict__ xin, const float* __restrict__ g,
                const float* __restrict__ gamma, const float* __restrict__ beta,
                float* __restrict__ xng) {
  const int wave = threadIdx.x >> 5;
  const int lane = threadIdx.x & 31;
  const int grp  = blockIdx.x * 8 + wave;   // over B*T*H groups
  const int h    = grp % H_;
  const size_t base = (size_t)grp * HS_;

  float x0 = xin[base + lane] * 0.125f;
  float x1 = xin[base + lane + 32] * 0.125f;
  float s  = x0 + x1;
  float s2 = x0 * x0 + x1 * x1;
#pragma unroll
  for (int off = 16; off >= 1; off >>= 1) {
    s  += __shfl_xor(s,  off, 32);
    s2 += __shfl_xor(s2, off, 32);
  }
  float mean = s * (1.f / HS_);
  float var  = s2 * (1.f / HS_) - mean * mean;
  float rstd = rsqrtf(var + EPS_);
  float y0 = (x0 - mean) * rstd * gamma[h * HS_ + lane]      + beta[h * HS_ + lane];
  float y1 = (x1 - mean) * rstd * gamma[h * HS_ + lane + 32] + beta[h * HS_ + lane + 32];
  xng[base + lane]      = y0 * g[base + lane];
  xng[base + lane + 32] = y1 * g[base + lane + 32];
}

// ---------------- host-side launch ----------------
extern "C" void kernel_launch(void* const* d_in, const int* in_sizes, int n_in,
                              void* d_out, int out_size, void* d_ws, size_t ws_size,
                              hipStream_t stream) {
  const float* x        = (const float*)d_in[0];
  const float* attstate = (const float*)d_in[1];
  const float* wkvstate = (const float*)d_in[2];
  const float* tmk      = (const float*)d_in[3];
  const float* tmv      = (const float*)d_in[4];
  const float* tmr      = (const float*)d_in[5];
  const float* tmg      = (const float*)d_in[6];
  const float* tdecay   = (const float*)d_in[7];
  const float* tfaaaa   = (const float*)d_in[8];
  const float* w_gate   = (const float*)d_in[9];
  const float* w_rec    = (const float*)d_in[10];
  const float* w_key    = (const float*)d_in[11];
  const float* w_val    = (const float*)d_in[12];
  const float* w_out    = (const float*)d_in[13];
  const float* gamma    = (const float*)d_in[14];
  const float* beta     = (const float*)d_in[15];

  float* y       = (float*)d_out;                     // B*T*C
  float* att_out = y + (size_t)B_ * T_ * C_;          // B*C
  float* S_out   = att_out + (size_t)B_ * C_;         // B*H*HS*HS

  const size_t n = (size_t)B_ * T_ * C_;
  float* ws   = (float*)d_ws;
  float* xk   = ws;
  float* xv   = ws + 1 * n;
  float* xr   = ws + 2 * n;
  float* xg   = ws + 3 * n;
  float* rbuf = ws + 4 * n;
  float* kbuf = ws + 5 * n;
  float* vbuf = ws + 6 * n;
  float* gbuf = ws + 7 * n;
  float* wkvo = ws + 8 * n;
  float* xng  = ws + 9 * n;
  float* wdec = ws + 10 * n;   // H*HS floats

  mix_kernel<<<(unsigned)((n + 255) / 256), 256, 0, stream>>>(
      x, attstate, tmk, tmv, tmr, tmg, xk, xv, xr, xg, att_out);
  wdecay_kernel<<<(H_ * HS_ + 255) / 256, 256, 0, stream>>>(tdecay, wdec);

  // batched projection GEMMs: r, k, v, g(silu) in a single launch (grid.z = 4)
  GemmBatch gb;
  gb.A[0] = xr; gb.W[0] = w_rec;  gb.O[0] = rbuf; gb.mode[0] = 0;
  gb.A[1] = xk; gb.W[1] = w_key;  gb.O[1] = kbuf; gb.mode[1] = 0;
  gb.A[2] = xv; gb.W[2] = w_val;  gb.O[2] = vbuf; gb.mode[2] = 0;
  gb.A[3] = xg; gb.W[3] = w_gate; gb.O[3] = gbuf; gb.mode[3] = 1;
  dim3 ggrid(C_ / BN, (B_ * T_) / BM, 4);
  gemm_wmma<<<ggrid, 256, 0, stream>>>(gb);

  wkv_scan<<<B_ * H_, 64, 0, stream>>>(rbuf, kbuf, vbuf, wdec, tfaaaa,
                                       wkvstate, wkvo, S_out);

  gnorm_gate<<<(B_ * T_ * H_) / 8, 256, 0, stream>>>(wkvo, gbuf, gamma, beta, xng);

  // final GEMM: y = (xn*g) @ w_output.T
  GemmBatch go;
  go.A[0] = xng; go.W[0] = w_out; go.O[0] = y; go.mode[0] = 0;
  go.A[1] = go.A[2] = go.A[3] = xng;
  go.W[1] = go.W[2] = go.W[3] = w_out;
  go.O[1] = go.O[2] = go.O[3] = y;
  go.mode[1] = go.mode[2] = go.mode[3] = 0;
  dim3 ogrid(C_ / BN, (B_ * T_) / BM, 1);
  gemm_wmma<<<ogrid, 256, 0, stream>>>(go);
}